// get_loss_76854144795249
// MI455X (gfx1250) — compile-verified
//
#include <hip/hip_runtime.h>
#include <hip/hip_bf16.h>

// CDNA5 (gfx1250) Chamfer distance using V_WMMA_F32_16X16X4_F32.
//
// Fold: A_m = (x0, x1, x2, 1), B_n = (-2*y0, -2*y1, -2*y2, ||y||^2)
//   =>  acc[m][n] = -2 * x_m . y_n + ||y_n||^2
// ||x_m||^2 is constant across the column sweep, so min-reduce the RAW
// accumulators and add it once at the end:  rowmin = max(0, min_n(acc) + x2).
// Inner loop per 16-col tile: 1x ds_load_b64 (B fragment direct, no selects),
// 1x WMMA, 8x fmin (-> min3 pairs). No pk_add, no cndmask, no clamp in loop.
//
// Each wave32 owns one 16-row strip of X and sweeps all N columns of Y
// (staged in LDS as float4 = {-2y0,-2y1,-2y2,||y||^2}). Two launches (roles
// swapped) give both Chamfer directions; per-strip sums -> workspace ->
// deterministic single-block reduce.

typedef __attribute__((ext_vector_type(2))) float v2f;
typedef __attribute__((ext_vector_type(8))) float v8f;

#define TILE 16
#define WAVES_PER_WG 4
#define WG_THREADS (WAVES_PER_WG * 32)
#define CHUNK 512  // Y points staged in LDS per step (8 KB as float4)

__global__ __launch_bounds__(WG_THREADS)
void chamfer_rowmin_kernel(const float* __restrict__ X,
                           const float* __restrict__ Y,
                           float* __restrict__ partials,
                           int N, int strips_per_batch)
{
    __shared__ float4 ybuf[CHUNK];

    const int b    = blockIdx.y;
    const int lane = threadIdx.x & 31;
    const int wave = threadIdx.x >> 5;
    const int strip = blockIdx.x * WAVES_PER_WG + wave;   // strip within batch
    const int rowBase = strip * TILE;

    const float* __restrict__ Xb = X + (size_t)b * N * 3;
    const float* __restrict__ Yb = Y + (size_t)b * N * 3;

    // ---- Load this wave's 16 X points and build the A fragment (16x4).
    // A layout: lanes 0-15 hold M=lane, VGPR0=K0, VGPR1=K1;
    //           lanes 16-31 hold M=lane-16, VGPR0=K2, VGPR1=K3 (=1.0).
    const int m = rowBase + (lane & 15);
    const float x0 = Xb[3 * m + 0];
    const float x1 = Xb[3 * m + 1];
    const float x2 = Xb[3 * m + 2];
    const float xnorm = x0 * x0 + x1 * x1 + x2 * x2;

    v2f afrag;
    afrag.x = (lane < 16) ? x0 : x2;
    afrag.y = (lane < 16) ? x1 : 1.0f;

    // ||x_m||^2 per accumulator VGPR i: C layout has
    // lanes 0-15 -> M = rowBase+i, lanes 16-31 -> M = rowBase+8+i.
    float x2v[8];
    const int srcOff = (lane >= 16) ? 8 : 0;
#pragma unroll
    for (int i = 0; i < 8; ++i)
        x2v[i] = __shfl(xnorm, i + srcOff, 32);

    float rmin[8];
#pragma unroll
    for (int i = 0; i < 8; ++i) rmin[i] = 3.402823466e38f;

    // B fragment loads: lane reads exactly its half of the staged float4.
    // float2 index = (point index)*2 + (1 if lane>=16): lanes 0-15 get
    // (-2y0,-2y1) = K0/K1, lanes 16-31 get (-2y2,||y||^2) = K2/K3.
    const float2* __restrict__ yb2 = reinterpret_cast<const float2*>(ybuf);
    const int fragIdx = ((lane & 15) << 1) | (lane >> 4);

    // ---- Sweep all N columns in LDS-staged chunks.
    for (int chunk = 0; chunk < N; chunk += CHUNK) {
        __syncthreads();  // previous chunk fully consumed
        for (int k = threadIdx.x; k < CHUNK; k += WG_THREADS) {
            const int n = chunk + k;
            const float y0 = Yb[3 * n + 0];
            const float y1 = Yb[3 * n + 1];
            const float y2 = Yb[3 * n + 2];
            ybuf[k] = make_float4(-2.0f * y0, -2.0f * y1, -2.0f * y2,
                                  y0 * y0 + y1 * y1 + y2 * y2);
        }
        __syncthreads();

#pragma unroll 4
        for (int t = 0; t < CHUNK; t += TILE) {
            const float2 p = yb2[(t << 1) + fragIdx];
            v2f bfrag;
            bfrag.x = p.x;
            bfrag.y = p.y;

            v8f cz = {};
            v8f acc = __builtin_amdgcn_wmma_f32_16x16x4_f32(
                /*neg_a=*/false, afrag, /*neg_b=*/false, bfrag,
                /*c_mod=*/(short)0, cz, /*reuse_a=*/false, /*reuse_b=*/false);

#pragma unroll
            for (int i = 0; i < 8; ++i)
                rmin[i] = fminf(rmin[i], acc[i]);
        }
    }

    // ---- Deferred ||x||^2 add + clamp, then cross-lane min within each
    // 16-lane half (same M, different N).
#pragma unroll
    for (int i = 0; i < 8; ++i) {
        rmin[i] = fmaxf(rmin[i] + x2v[i], 0.0f);
#pragma unroll
        for (int off = 1; off < 16; off <<= 1)
            rmin[i] = fminf(rmin[i], __shfl_xor(rmin[i], off, 32));
    }

    // Sum the 16 row-mins of this strip (8 in each half), combine halves.
    float s = 0.0f;
#pragma unroll
    for (int i = 0; i < 8; ++i) s += rmin[i];
    s += __shfl_xor(s, 16, 32);

    if (lane == 0)
        partials[(size_t)b * strips_per_batch + strip] = s;
}

__global__ __launch_bounds__(256)
void finalize_kernel(const float* __restrict__ partials, int nPartials,
                     const float* __restrict__ fbpp, const int* __restrict__ lam,
                     float* __restrict__ out, int B, int N)
{
    __shared__ float red[256];
    float s = 0.0f;
    for (int i = threadIdx.x; i < nPartials; i += 256) s += partials[i];
    red[threadIdx.x] = s;
    __syncthreads();
    for (int w = 128; w > 0; w >>= 1) {
        if ((int)threadIdx.x < w) red[threadIdx.x] += red[threadIdx.x + w];
        __syncthreads();
    }
    if (threadIdx.x == 0) {
        const float cham = red[0] / (float)((size_t)B * (size_t)N);
        float r = 0.0f;
        for (int i = 0; i < B; ++i) r += fbpp[i];
        r /= (float)B;
        out[0] = cham + (float)lam[0] * r;
    }
}

extern "C" void kernel_launch(void* const* d_in, const int* in_sizes, int n_in,
                              void* d_out, int out_size, void* d_ws, size_t ws_size,
                              hipStream_t stream)
{
    const float* pred = (const float*)d_in[0];   // (B, N, 3) f32
    const float* targ = (const float*)d_in[1];   // (B, N, 3) f32
    const float* fbpp = (const float*)d_in[2];   // (B,) f32
    const int*   lam  = (const int*)d_in[3];     // scalar int

    const int B = in_sizes[2];
    const int N = in_sizes[0] / (3 * B);         // 8192
    const int strips = N / TILE;                 // 512 per batch

    float* partials = (float*)d_ws;              // 2*B*strips floats (16 KB)

    dim3 grid(strips / WAVES_PER_WG, B, 1);
    // Pass 1: rows = pred, cols = target  -> cham_x row-mins
    chamfer_rowmin_kernel<<<grid, WG_THREADS, 0, stream>>>(
        pred, targ, partials, N, strips);
    // Pass 2: rows = target, cols = pred  -> cham_y row-mins
    chamfer_rowmin_kernel<<<grid, WG_THREADS, 0, stream>>>(
        targ, pred, partials + (size_t)B * strips, N, strips);

    finalize_kernel<<<1, 256, 0, stream>>>(
        partials, 2 * B * strips, fbpp, lam, (float*)d_out, B, N);
}